// ChampherLoss_53042846105607
// MI455X (gfx1250) — compile-verified
//
#include <hip/hip_runtime.h>

typedef float v2f __attribute__((ext_vector_type(2)));
typedef float v8f __attribute__((ext_vector_type(8)));

// Monotone float->uint encoding so unsigned atomicMin orders like float min.
__device__ __forceinline__ unsigned enc_f32(float f) {
    unsigned b = __float_as_uint(f);
    unsigned mask = (unsigned)(((int)b) >> 31) | 0x80000000u;
    return b ^ mask;
}
__device__ __forceinline__ float dec_f32(unsigned u) {
    unsigned b = (u & 0x80000000u) ? (u ^ 0x80000000u) : ~u;
    return __uint_as_float(b);
}

__global__ void chamfer_init_out(float* out) {
    out[0] = 0.0f;
}

// One block per (B*P) batch. X is "reconstructed_points" (M=1024 pts, with the
// reference's faithful permute(0,1,3,2).reshape(-1,M,3) scramble), Y is
// "normalized_receptive_fields" (N=1024 pts, contiguous).
__global__ __launch_bounds__(256) void chamfer_kernel(const float* __restrict__ Y,
                                                      const float* __restrict__ X,
                                                      float* __restrict__ out) {
    constexpr int M = 1024, N = 1024;
    // Stride-4 padded coords: slot 3 is 0.0f so the K=3 WMMA lane reads zero
    // with a plain ds_load_b64 (no divergent conditional in the hot loop).
    __shared__ float xs4[4 * M];     // xs4[m*4 + c], c in 0..3 (c==3 -> 0)
    __shared__ float ys4[4 * N];     // ys4[n*4 + c]
    __shared__ float xx[M];          // ||x_m||^2
    __shared__ float yy[N];          // ||y_n||^2
    __shared__ float rowmin[M];      // min over n per m (exclusive writes)
    __shared__ unsigned colmin[N];   // encoded min over m per n (atomic)
    __shared__ float red[256];

    const int t  = threadIdx.x;
    const int bp = blockIdx.x;
    const float* Xb = X + (size_t)bp * M * 3;
    const float* Yb = Y + (size_t)bp * N * 3;

    // Stage inputs into LDS. Faithful scramble: x[m][c] lives at flat g=3m+c of
    // the (3,M) coordinate-major view => X[(g & 1023)*3 + (g >> 10)].
    for (int i = t; i < 4 * M; i += 256) {
        const int m = i >> 2, c = i & 3;
        float xv = 0.0f, yv = 0.0f;
        if (c < 3) {
            const int g = 3 * m + c;
            xv = Xb[(g & 1023) * 3 + (g >> 10)];
            yv = Yb[g];
        }
        xs4[i] = xv;
        ys4[i] = yv;
    }
    for (int i = t; i < N; i += 256) colmin[i] = 0xFFFFFFFFu;
    __syncthreads();
    for (int i = t; i < M; i += 256) {
        float a0 = xs4[4 * i], a1 = xs4[4 * i + 1], a2 = xs4[4 * i + 2];
        xx[i] = a0 * a0 + a1 * a1 + a2 * a2;
        float b0 = ys4[4 * i], b1 = ys4[4 * i + 1], b2 = ys4[4 * i + 2];
        yy[i] = b0 * b0 + b1 * b1 + b2 * b2;
    }
    __syncthreads();

    const int lane = t & 31;          // wave32
    const int wave = t >> 5;          // 8 waves
    const int half = lane >> 4;       // 0: lanes 0-15, 1: lanes 16-31
    const int l16  = lane & 15;

    // Each wave owns m-tiles {wave, wave+8, ...}: 8 tiles of 16 rows.
    for (int mt = wave; mt < M / 16; mt += 8) {
        const int m_base = mt << 4;
        // A fragment, 16x4 f32: lane half selects K pair {0,1} or {2,3};
        // K=3 reads the zero pad slot. Single aligned ds_load_b64.
        const v2f a = *(const v2f*)&xs4[((m_base + l16) << 2) + (half << 1)];
        float xxv[8];
#pragma unroll
        for (int r = 0; r < 8; ++r) xxv[r] = xx[m_base + (half << 3) + r];

        v8f rmin;
#pragma unroll
        for (int r = 0; r < 8; ++r) rmin[r] = __builtin_inff();

        for (int nt = 0; nt < N / 16; ++nt) {
            const int n_base = nt << 4;
            // B fragment, 4x16 f32: symmetric layout, N = lane&15.
            const v2f b = *(const v2f*)&ys4[((n_base + l16) << 2) + (half << 1)];
            v8f c = {};
            // xy tile: D = A x B (+0), 16x16 f32 accumulator across 8 VGPRs.
            v8f acc = __builtin_amdgcn_wmma_f32_16x16x4_f32(
                false, a, false, b, (short)0, c, false, false);

            const float yyv = yy[n_base + l16];
            float cmin = __builtin_inff();
#pragma unroll
            for (int r = 0; r < 8; ++r) {
                // C/D layout: VGPR r holds M = m_base + half*8 + r, N = n_base + l16.
                float d = xxv[r] + yyv - 2.0f * acc[r];
                rmin[r] = fminf(rmin[r], d);
                cmin = fminf(cmin, d);
            }
            // Combine the two halves (different M rows, same N column).
            cmin = fminf(cmin, __shfl_xor(cmin, 16));
            atomicMin(&colmin[n_base + l16], enc_f32(cmin));
        }

        // Reduce row minima across the 16 columns held in each 16-lane half.
#pragma unroll
        for (int r = 0; r < 8; ++r) {
            float v = rmin[r];
            v = fminf(v, __shfl_xor(v, 1));
            v = fminf(v, __shfl_xor(v, 2));
            v = fminf(v, __shfl_xor(v, 4));
            v = fminf(v, __shfl_xor(v, 8));
            if (l16 == 0) rowmin[m_base + (half << 3) + r] = v;
        }
    }
    __syncthreads();

    // Block sums of row/col minima, pre-scaled to contribute to the two means.
    float rs = 0.0f, cs = 0.0f;
    for (int i = t; i < M; i += 256) {
        rs += rowmin[i];
        cs += dec_f32(colmin[i]);
    }
    const float inv_bpm = 1.0f / (256.0f * 1024.0f);  // BP*M == BP*N
    red[t] = rs * inv_bpm + cs * inv_bpm;
    __syncthreads();
    for (int s = 128; s > 0; s >>= 1) {
        if (t < s) red[t] += red[t + s];
        __syncthreads();
    }
    if (t == 0) atomicAdd(out, red[0]);
}

extern "C" void kernel_launch(void* const* d_in, const int* in_sizes, int n_in,
                              void* d_out, int out_size, void* d_ws, size_t ws_size,
                              hipStream_t stream) {
    const float* Y = (const float*)d_in[0];  // normalized_receptive_fields (B,P,N,3)
    const float* X = (const float*)d_in[1];  // reconstructed_points (B,P,M,3)
    float* out = (float*)d_out;

    chamfer_init_out<<<1, 1, 0, stream>>>(out);
    chamfer_kernel<<<256, 256, 0, stream>>>(Y, X, out);
}